// NonLocalAggregation_59193239274145
// MI455X (gfx1250) — compile-verified
//
#include <hip/hip_runtime.h>

typedef __attribute__((ext_vector_type(2))) float v2f;
typedef __attribute__((ext_vector_type(8))) float v8f;

#define BATCH 32
#define FDIM  64
#define NPTS  1024
#define KNN   8
#define OUTC  64

// ---------------------------------------------------------------------------
// Kernel 0: squared norms sq[b,n] = sum_f x[b,f,n]^2
// ---------------------------------------------------------------------------
__global__ __launch_bounds__(256)
void nla_sqnorm(const float* __restrict__ x, float* __restrict__ sq) {
  int t = blockIdx.x * 256 + threadIdx.x;      // t = b*NPTS + n
  int b = t >> 10, n = t & 1023;
  const float* xp = x + (size_t)b * FDIM * NPTS + n;
  float acc = 0.f;
#pragma unroll
  for (int f = 0; f < FDIM; ++f) { float v = xp[f * NPTS]; acc += v * v; }
  sq[t] = acc;
}

// ---------------------------------------------------------------------------
// Kernel 1: fused distance Gram (f32 WMMA) + top-8 selection.
// Block = (row-tile of 16 queries, batch). 256 threads = 8 waves.
// Each wave computes 8 column tiles of 16x16 dots via chained
// v_wmma_f32_16x16x4_f32, converts to ranking keys (sq_j - 2*dot) in LDS,
// then each wave selects top-8 smallest for 2 rows via shfl argmin merge.
// ---------------------------------------------------------------------------
__global__ __launch_bounds__(256)
void nla_knn(const float* __restrict__ x, const float* __restrict__ sq,
             int* __restrict__ idxout) {
  __shared__ float keys[16 * 1028];            // padded stride: bank-friendly
  const int b    = blockIdx.y;
  const int r0   = blockIdx.x * 16;
  const int lane = threadIdx.x & 31;
  const int wave = threadIdx.x >> 5;
  const int half = lane >> 4;                  // 0: lanes 0-15, 1: lanes 16-31
  const int l16  = lane & 15;
  const float* xb = x + (size_t)b * FDIM * NPTS;

  // A fragments for this block's 16 query rows (K = 64 -> 16 frags of K=4).
  // f32 16x4 A layout: lanes 0-15 hold K = kb,kb+1 ; lanes 16-31 K = kb+2,kb+3
  v2f afr[16];
  {
    int row = r0 + l16;
#pragma unroll
    for (int kk = 0; kk < 16; ++kk) {
      int kb = kk * 4 + half * 2;
      afr[kk].x = xb[(size_t)kb * NPTS + row];
      afr[kk].y = xb[(size_t)(kb + 1) * NPTS + row];
    }
  }

  // Phase 1: 64 column tiles, wave w owns tiles {w, w+8, ..., w+56}
  for (int t8 = 0; t8 < 8; ++t8) {
    int c0  = (wave + t8 * 8) * 16;
    int col = c0 + l16;
    v2f bfr[16];
#pragma unroll
    for (int kk = 0; kk < 16; ++kk) {
      int kb = kk * 4 + half * 2;
      bfr[kk].x = xb[(size_t)kb * NPTS + col];
      bfr[kk].y = xb[(size_t)(kb + 1) * NPTS + col];
    }
    v8f acc = {};
#pragma unroll
    for (int kk = 0; kk < 16; ++kk) {
      acc = __builtin_amdgcn_wmma_f32_16x16x4_f32(
          false, afr[kk], false, bfr[kk], (short)0, acc, false, false);
    }
    // key_j = sq_j - 2*dot(i,j)  (equals d2 - sq_i: same ranking per row i)
    float sqc = sq[b * NPTS + col];
#pragma unroll
    for (int v = 0; v < 8; ++v) {
      int rm = v + half * 8;                   // C/D layout row
      keys[rm * 1028 + col] = sqc - 2.0f * acc[v];
    }
  }
  __syncthreads();

  // Phase 2: top-8 smallest per row. Wave w handles rows {w, w+8}.
  for (int rr = 0; rr < 2; ++rr) {
    int row = wave + rr * 8;
    float tv[KNN]; int ti[KNN];
#pragma unroll
    for (int s = 0; s < KNN; ++s) { tv[s] = 3.4e38f; ti[s] = 0x7fffffff; }
    // each lane scans 32 strided candidates, keeps private top-8
    for (int j = lane; j < NPTS; j += 32) {
      float kv = keys[row * 1028 + j];
      int ms = 0; float mv = tv[0];
#pragma unroll
      for (int s = 1; s < KNN; ++s) if (tv[s] > mv) { mv = tv[s]; ms = s; }
      if (kv < mv) { tv[ms] = kv; ti[ms] = j; }
    }
    // 8 rounds of wave32 argmin extraction
    int res[KNN];
    for (int r8 = 0; r8 < KNN; ++r8) {
      float bv = 3.4e38f; int bi = 0x7fffffff;
#pragma unroll
      for (int s = 0; s < KNN; ++s) if (tv[s] < bv) { bv = tv[s]; bi = ti[s]; }
      float v = bv; int gi = bi;
#pragma unroll
      for (int off = 16; off > 0; off >>= 1) {
        float ov = __shfl_xor(v, off, 32);
        int   og = __shfl_xor(gi, off, 32);
        if (ov < v || (ov == v && og < gi)) { v = ov; gi = og; }
      }
      res[r8] = gi;
#pragma unroll
      for (int s = 0; s < KNN; ++s)
        if (ti[s] == gi) { tv[s] = 3.4e38f; ti[s] = 0x7fffffff; }
    }
    if (lane < KNN)
      idxout[((size_t)b * NPTS + (r0 + row)) * KNN + lane] = res[lane];
  }
}

// ---------------------------------------------------------------------------
// Kernel 2: gather neighbor mean, then fused K=128 GEMM via f32 WMMA:
//   out = [meanNeigh - xf | xf] @ [Wd ; Ws]^T + (bd + bs + bias)
// Block = (16 rows, batch). 128 threads = 4 waves; wave w -> out cols w*16..
// ---------------------------------------------------------------------------
__global__ __launch_bounds__(128)
void nla_agg(const float* __restrict__ x, const int* __restrict__ idxin,
             const float* __restrict__ Wd, const float* __restrict__ bd,
             const float* __restrict__ Ws, const float* __restrict__ bs,
             const float* __restrict__ bias, float* __restrict__ out) {
  __shared__ float amat[16 * 132];             // 16 rows x 128 (padded)
  const int b  = blockIdx.y;
  const int r0 = blockIdx.x * 16;
  const float* xb = x + (size_t)b * FDIM * NPTS;

  // Phase 1: build A = [meanNeigh - xf | xf] in LDS
  for (int e = threadIdx.x; e < 16 * FDIM; e += 128) {
    int r = e >> 6, f = e & 63;
    int row = r0 + r;
    const int* mi = idxin + ((size_t)b * NPTS + row) * KNN;
    float s = 0.f;
#pragma unroll
    for (int j = 0; j < KNN; ++j) s += xb[(size_t)f * NPTS + mi[j]];
    float own = xb[(size_t)f * NPTS + row];
    amat[r * 132 + f]        = s * 0.125f - own;  // mean(neigh) - self
    amat[r * 132 + 64 + f]   = own;
  }
  __syncthreads();

  // Phase 2: 16x16 output tile per wave, K = 128 -> 32 chained f32 WMMAs
  const int lane = threadIdx.x & 31;
  const int wave = threadIdx.x >> 5;
  const int half = lane >> 4;
  const int l16  = lane & 15;
  const int o    = wave * 16 + l16;            // output channel for B frag
  v8f acc = {};
#pragma unroll
  for (int kk = 0; kk < 32; ++kk) {
    int kb = kk * 4 + half * 2;
    v2f av, bv;
    av.x = amat[l16 * 132 + kb];
    av.y = amat[l16 * 132 + kb + 1];
    const float* wp = (kb < 64) ? (Wd + o * 64 + kb)
                                : (Ws + o * 64 + (kb - 64));
    bv.x = wp[0];
    bv.y = wp[1];
    acc = __builtin_amdgcn_wmma_f32_16x16x4_f32(
        false, av, false, bv, (short)0, acc, false, false);
  }
  float addc = bd[o] + bs[o] + bias[o];
#pragma unroll
  for (int v = 0; v < 8; ++v) {
    int rm = v + half * 8;
    out[(size_t)b * OUTC * NPTS + (size_t)o * NPTS + (r0 + rm)] = acc[v] + addc;
  }
}

// ---------------------------------------------------------------------------
extern "C" void kernel_launch(void* const* d_in, const int* in_sizes, int n_in,
                              void* d_out, int out_size, void* d_ws, size_t ws_size,
                              hipStream_t stream) {
  const float* x    = (const float*)d_in[0];
  const float* Wd   = (const float*)d_in[1];
  const float* bd   = (const float*)d_in[2];
  const float* Ws   = (const float*)d_in[3];
  const float* bs   = (const float*)d_in[4];
  const float* bias = (const float*)d_in[5];
  float* out = (float*)d_out;

  // workspace: sq (B*N f32) then idx (B*N*K i32)
  float* sq  = (float*)d_ws;
  int*   idx = (int*)((char*)d_ws + (size_t)BATCH * NPTS * sizeof(float));

  nla_sqnorm<<<(BATCH * NPTS) / 256, 256, 0, stream>>>(x, sq);
  nla_knn<<<dim3(NPTS / 16, BATCH), 256, 0, stream>>>(x, sq, idx);
  nla_agg<<<dim3(NPTS / 16, BATCH), 128, 0, stream>>>(x, idx, Wd, bd, Ws, bs,
                                                      bias, out);
}